// OCGatherEnergyCorrFac_new_81235011436601
// MI455X (gfx1250) — compile-verified
//
#include <hip/hip_runtime.h>
#include <stdint.h>

// ---------------------------------------------------------------------------
// OCGatherEnergyCorrFac for MI455X (gfx1250).
//
//   K0: zero 3 segment accumulators in d_ws (re-zeroed every call).
//   K1: scatter — stream the 4 per-hit inputs through LDS with gfx1250
//       async global->LDS b128 loads (512B/wave/issue, double buffered,
//       s_wait_asynccnt synchronized), then conditional
//       global_atomic_add_f32 into L2-resident segment arrays (~5M atomics).
//   K2: per-segment — alpha-index corrections, precompute all five
//       per-segment outputs (incl. minbias fraction).
//   K3: gather — broadcast 5 L2-resident tables back to all hits,
//       4 hits/thread with b128 stores.
//
// HBM traffic ~160 MB total -> ~7 us floor at 23.3 TB/s.
// Workspace: 8 float arrays of NSEG=20001 (~640 KB).
// ---------------------------------------------------------------------------

__device__ __forceinline__ uint32_t lds_off(const void* p) {
  // Generic LDS address keeps the LDS byte offset in bits[31:0].
  return (uint32_t)(uintptr_t)p;
}

__device__ __forceinline__ void async_ld_b128(uint32_t lds, uint32_t voff,
                                              const void* sbase) {
  // GVS addressing: mem = SGPR64 + VGPR32 byte offset; VDST = LDS address.
  asm volatile("global_load_async_to_lds_b128 %0, %1, %2"
               :: "v"(lds), "v"(voff), "s"(sbase)
               : "memory");
}

__global__ void __launch_bounds__(256) oc_zero_kernel(float* p, int n) {
  int i = blockIdx.x * blockDim.x + threadIdx.x;
  if (i < n) p[i] = 0.0f;
}

#define CHUNK 128  // elements per wave-chunk (32 lanes x 4 elements)

__global__ void __launch_bounds__(256) oc_scatter_kernel(
    const int* __restrict__ pred_sid, const float* __restrict__ energy,
    const int* __restrict__ is_track, const int* __restrict__ t_mb,
    float* __restrict__ seg_hit, float* __restrict__ seg_trk,
    float* __restrict__ seg_nmb, int N) {
  // Per-block: 8 waves x 2 buffers x 4 arrays x 128 elems x 4B = 32 KB.
  __shared__ int   ls_sid[2][8][CHUNK];
  __shared__ float ls_e  [2][8][CHUNK];
  __shared__ int   ls_tk [2][8][CHUNK];
  __shared__ int   ls_mb [2][8][CHUNK];

  const int tid    = threadIdx.x;
  const int lane   = tid & 31;
  const int w      = tid >> 5;  // wave in block
  const int gWave  = (int)(blockIdx.x * (blockDim.x >> 5)) + w;
  const int nWaves = (int)(gridDim.x * (blockDim.x >> 5));
  const int NC     = N / CHUNK;  // full chunks

  auto issue = [&](int c, int b) {
    // lane covers elements c*128 + 4*lane .. +3 (16B contiguous per lane).
    uint32_t off = (uint32_t)c * (CHUNK * 4u) + (uint32_t)lane * 16u;
    async_ld_b128(lds_off(&ls_sid[b][w][lane * 4]), off, pred_sid);
    async_ld_b128(lds_off(&ls_e  [b][w][lane * 4]), off, energy);
    async_ld_b128(lds_off(&ls_tk [b][w][lane * 4]), off, is_track);
    async_ld_b128(lds_off(&ls_mb [b][w][lane * 4]), off, t_mb);
  };

  auto accumulate = [&](int sid, float e, int tk, int mb) {
    if (sid < 0) e = 0.0f;  // noise hits contribute nothing
    float eh = (tk == 0) ? e : 0.0f;
    float et = (tk == 1) ? e : 0.0f;
    float en = (mb == 1) ? 0.0f : eh;
    int   s  = sid + 1;  // [0, NSEG-1]
    if (eh != 0.0f) atomicAdd(&seg_hit[s], eh);
    if (et != 0.0f) atomicAdd(&seg_trk[s], et);
    if (en != 0.0f) atomicAdd(&seg_nmb[s], en);
  };

  int c = gWave, buf = 0;
  if (c < NC) issue(c, buf);
  while (c < NC) {
    int next = c + nWaves;
    if (next < NC) {
      issue(next, buf ^ 1);
      // 4 new loads on top of <=4 pending; async loads complete in order, so
      // ASYNCcnt<=4 means the current buffer's 4 loads are done.
      asm volatile("s_wait_asynccnt 4" ::: "memory");
    } else {
      asm volatile("s_wait_asynccnt 0" ::: "memory");
    }
    // Read back this lane's 4 staged elements (conflict-free b128 pattern).
    int4   s4 = *(const int4*)  &ls_sid[buf][w][lane * 4];
    float4 e4 = *(const float4*)&ls_e  [buf][w][lane * 4];
    int4   t4 = *(const int4*)  &ls_tk [buf][w][lane * 4];
    int4   m4 = *(const int4*)  &ls_mb [buf][w][lane * 4];
    accumulate(s4.x, e4.x, t4.x, m4.x);
    accumulate(s4.y, e4.y, t4.y, m4.y);
    accumulate(s4.z, e4.z, t4.z, m4.z);
    accumulate(s4.w, e4.w, t4.w, m4.w);
    c = next;
    buf ^= 1;
  }

  // Tail (N % 128) via direct loads — block 0 only.
  int tail = N % CHUNK;
  if (blockIdx.x == 0 && tid < tail) {
    int i = (N - tail) + tid;
    accumulate(pred_sid[i], energy[i], is_track[i], t_mb[i]);
  }
}

__global__ void __launch_bounds__(256) oc_percluster_kernel(
    const int* __restrict__ pred_sid, const float* __restrict__ corrf,
    const int* __restrict__ alpha_t, const int* __restrict__ alpha_h,
    const float* __restrict__ seg_hit, const float* __restrict__ seg_trk,
    const float* __restrict__ seg_nmb,
    float* __restrict__ o_traw, float* __restrict__ o_tcor,
    float* __restrict__ o_hraw, float* __restrict__ o_hcor,
    float* __restrict__ o_frac, int NSEG) {
  int s = blockIdx.x * blockDim.x + threadIdx.x;
  if (s >= NSEG) return;
  float cH = 0.0f, cT = 0.0f;
  if (s > 0) {
    // idx == -1 wraps to the zero appendix in the reference -> correction 0;
    // corr is also zeroed for noise hits (pred_sid == -1).
    int ih = alpha_h[s - 1];
    if (ih >= 0) cH = (pred_sid[ih] == -1) ? 0.0f : corrf[ih];
    int it = alpha_t[s - 1];
    if (it >= 0) cT = (pred_sid[it] == -1) ? 0.0f : corrf[it];
  }
  float hr = seg_hit[s], tr = seg_trk[s], nm = seg_nmb[s];
  o_traw[s] = tr;
  o_tcor[s] = tr * cT;
  o_hraw[s] = hr;
  o_hcor[s] = hr * cH;
  o_frac[s] = (hr == 0.0f) ? 0.0f : nm / hr;
}

// Fast path: 4 hits/thread, b128 stores. Requires N % 4 == 0.
__global__ void __launch_bounds__(256) oc_gather4_kernel(
    const int4* __restrict__ sid4,
    const float* __restrict__ o_traw, const float* __restrict__ o_tcor,
    const float* __restrict__ o_hraw, const float* __restrict__ o_hcor,
    const float* __restrict__ o_frac,
    float* __restrict__ out, int N4) {
  int i = blockIdx.x * blockDim.x + threadIdx.x;
  if (i >= N4) return;
  int4 s = sid4[i];
  s.x += 1; s.y += 1; s.z += 1; s.w += 1;
  size_t n4 = (size_t)N4;  // rows are N = 4*N4 floats; float4 rows are N4.
  float4* o = (float4*)out;
  o[i]          = make_float4(o_traw[s.x], o_traw[s.y], o_traw[s.z], o_traw[s.w]);
  o[n4 + i]     = make_float4(o_tcor[s.x], o_tcor[s.y], o_tcor[s.z], o_tcor[s.w]);
  o[2 * n4 + i] = make_float4(o_hraw[s.x], o_hraw[s.y], o_hraw[s.z], o_hraw[s.w]);
  o[3 * n4 + i] = make_float4(o_hcor[s.x], o_hcor[s.y], o_hcor[s.z], o_hcor[s.w]);
  o[4 * n4 + i] = make_float4(o_frac[s.x], o_frac[s.y], o_frac[s.z], o_frac[s.w]);
}

// General fallback: 1 hit/thread.
__global__ void __launch_bounds__(256) oc_gather_kernel(
    const int* __restrict__ pred_sid,
    const float* __restrict__ o_traw, const float* __restrict__ o_tcor,
    const float* __restrict__ o_hraw, const float* __restrict__ o_hcor,
    const float* __restrict__ o_frac,
    float* __restrict__ out, int N) {
  int i = blockIdx.x * blockDim.x + threadIdx.x;
  if (i >= N) return;
  int s = pred_sid[i] + 1;
  size_t n = (size_t)N;
  out[i]         = o_traw[s];
  out[n + i]     = o_tcor[s];
  out[2 * n + i] = o_hraw[s];
  out[3 * n + i] = o_hcor[s];
  out[4 * n + i] = o_frac[s];
}

extern "C" void kernel_launch(void* const* d_in, const int* in_sizes, int n_in,
                              void* d_out, int out_size, void* d_ws, size_t ws_size,
                              hipStream_t stream) {
  const int*   pred_sid = (const int*)d_in[0];
  const float* corrf    = (const float*)d_in[1];
  const float* energy   = (const float*)d_in[2];
  // d_in[3] (no_noise_idx) and d_in[4] (pred_beta) are unused by the reference.
  const int*   is_track = (const int*)d_in[5];
  const int*   alpha_t  = (const int*)d_in[6];
  const int*   alpha_h  = (const int*)d_in[7];
  const int*   t_mb     = (const int*)d_in[8];

  const int N    = in_sizes[0];
  const int S    = in_sizes[7];  // alpha_idx_hits length
  const int NSEG = S + 1;

  float* ws      = (float*)d_ws;
  float* seg_hit = ws + 0 * NSEG;
  float* seg_trk = ws + 1 * NSEG;
  float* seg_nmb = ws + 2 * NSEG;
  float* o_traw  = ws + 3 * NSEG;
  float* o_tcor  = ws + 4 * NSEG;
  float* o_hraw  = ws + 5 * NSEG;
  float* o_hcor  = ws + 6 * NSEG;
  float* o_frac  = ws + 7 * NSEG;

  int z = 3 * NSEG;
  oc_zero_kernel<<<(z + 255) / 256, 256, 0, stream>>>(seg_hit, z);

  oc_scatter_kernel<<<768, 256, 0, stream>>>(
      pred_sid, energy, is_track, t_mb, seg_hit, seg_trk, seg_nmb, N);

  oc_percluster_kernel<<<(NSEG + 255) / 256, 256, 0, stream>>>(
      pred_sid, corrf, alpha_t, alpha_h, seg_hit, seg_trk, seg_nmb,
      o_traw, o_tcor, o_hraw, o_hcor, o_frac, NSEG);

  if ((N & 3) == 0) {
    int N4 = N >> 2;
    oc_gather4_kernel<<<(N4 + 255) / 256, 256, 0, stream>>>(
        (const int4*)pred_sid, o_traw, o_tcor, o_hraw, o_hcor, o_frac,
        (float*)d_out, N4);
  } else {
    oc_gather_kernel<<<(N + 255) / 256, 256, 0, stream>>>(
        pred_sid, o_traw, o_tcor, o_hraw, o_hcor, o_frac, (float*)d_out, N);
  }
}